// GATCell_50276887167262
// MI455X (gfx1250) — compile-verified
//
#include <hip/hip_runtime.h>

typedef __attribute__((ext_vector_type(2))) float v2f;
typedef __attribute__((ext_vector_type(8))) float v8f;

#define N_NODES   100000
#define N_EDGES   1600000
#define IN_DIM    128
#define OUT_DIM   32
#define HEADS     8
#define HF        256              // HEADS*OUT_DIM
#define E_TOT     (N_EDGES + N_NODES)   // edges + self loops
#define NEG_SLOPE 0.2f

// ---------------------------------------------------------------------------
// Init: m = -inf, denom = 0, out = bias (mean+bias epilogue folded into init)
// ---------------------------------------------------------------------------
__global__ __launch_bounds__(256) void init_bufs(float* __restrict__ m,
                                                 float* __restrict__ denom,
                                                 float* __restrict__ out,
                                                 const float* __restrict__ bias)
{
    int id = blockIdx.x * 256 + threadIdx.x;
    if (id < N_NODES * HEADS) {
        m[id]     = -__builtin_inff();
        denom[id] = 0.0f;
    }
    if (id < N_NODES * OUT_DIM) {
        out[id] = bias[id & (OUT_DIM - 1)];
    }
}

// ---------------------------------------------------------------------------
// GEMM: h[N,256] = X[N,128] @ W[128,256] via V_WMMA_F32_16X16X4_F32
// One block = 16 rows of X staged in LDS (padded stride 132 -> conflict-free),
// 8 waves each own two 16-wide column tiles, K-loop of 32 WMMAs per tile.
// ---------------------------------------------------------------------------
__global__ __launch_bounds__(256) void gemm_h(const float* __restrict__ X,
                                              const float* __restrict__ W,
                                              float* __restrict__ h)
{
    __shared__ float Xs[16 * 132];
    const int m0  = blockIdx.x * 16;
    const int tid = threadIdx.x;

    // cooperative, coalesced load of the 16x128 X tile
    for (int i = tid; i < 16 * 128; i += 256) {
        int r = i >> 7, c = i & 127;
        Xs[r * 132 + c] = X[(size_t)(m0 + r) * IN_DIM + c];
    }
    __syncthreads();

    const int wave = tid >> 5;
    const int lane = tid & 31;
    const int hi   = lane >> 4;       // 0: K pair (k,k+1)   1: K pair (k+2,k+3)
    const int l16  = lane & 15;
    const int n0   = wave * 32;       // two col tiles: n0 and n0+16

    v8f c0 = {0.f, 0.f, 0.f, 0.f, 0.f, 0.f, 0.f, 0.f};
    v8f c1 = c0;

    for (int k = 0; k < IN_DIM; k += 4) {
        // A fragment (16x4 f32): lane<16 -> K=k,k+1 ; lane>=16 -> K=k+2,k+3
        const float* ap = &Xs[l16 * 132 + k + 2 * hi];
        v2f a; a.x = ap[0]; a.y = ap[1];

        // B fragments (4x16 f32), mirrored layout
        const int krow = k + 2 * hi;
        v2f b0, b1;
        b0.x = W[(size_t)krow       * HF + n0 + l16];
        b0.y = W[(size_t)(krow + 1) * HF + n0 + l16];
        b1.x = W[(size_t)krow       * HF + n0 + 16 + l16];
        b1.y = W[(size_t)(krow + 1) * HF + n0 + 16 + l16];

        c0 = __builtin_amdgcn_wmma_f32_16x16x4_f32(false, a, false, b0,
                                                   (short)0, c0, false, false);
        c1 = __builtin_amdgcn_wmma_f32_16x16x4_f32(false, a, false, b1,
                                                   (short)0, c1, false, false);
    }

    // C/D layout: VGPR r = row (r + 8*hi), lanes give N = n0 + l16
#pragma unroll
    for (int r = 0; r < 8; ++r) {
        const size_t row = (size_t)(m0 + r + hi * 8);
        h[row * HF + n0 + l16]      = c0[r];
        h[row * HF + n0 + 16 + l16] = c1[r];
    }
}

// ---------------------------------------------------------------------------
// Per-(node,head) attention logits: a_src = <h, att_src>, a_dst = <h, att_dst>
// ---------------------------------------------------------------------------
__global__ __launch_bounds__(256) void attn_logits(const float* __restrict__ h,
                                                   const float* __restrict__ att_src,
                                                   const float* __restrict__ att_dst,
                                                   float* __restrict__ a_src,
                                                   float* __restrict__ a_dst)
{
    int id = blockIdx.x * 256 + threadIdx.x;
    if (id >= N_NODES * HEADS) return;
    const int n = id >> 3, hd = id & 7;
    const float4* hp = (const float4*)&h[(size_t)n * HF + hd * OUT_DIM];
    const float4* sp = (const float4*)&att_src[hd * OUT_DIM];
    const float4* dp = (const float4*)&att_dst[hd * OUT_DIM];
    float s = 0.f, d = 0.f;
#pragma unroll
    for (int i = 0; i < OUT_DIM / 4; ++i) {
        float4 hv = hp[i], sv = sp[i], dv = dp[i];
        s += hv.x * sv.x + hv.y * sv.y + hv.z * sv.z + hv.w * sv.w;
        d += hv.x * dv.x + hv.y * dv.y + hv.z * dv.z + hv.w * dv.w;
    }
    a_src[id] = s;
    a_dst[id] = d;
}

// ---------------------------------------------------------------------------
// Edge helpers
// ---------------------------------------------------------------------------
__device__ __forceinline__ void edge_sd(const int* __restrict__ ei, int e,
                                        int& s, int& d)
{
    if (e < N_EDGES) { s = ei[e]; d = ei[N_EDGES + e]; }
    else             { s = d = e - N_EDGES; }        // appended self loop
}

__device__ __forceinline__ float lrelu(float x)
{
    return x >= 0.f ? x : NEG_SLOPE * x;
}

// float atomic max with monotonic int/uint encoding (init must be -inf):
// non-negative floats: int ordering == float ordering -> atomicMax(int)
// negative floats: uint ordering reversed -> atomicMin(uint); a negative-path
// min can never displace a stored non-negative (smaller uint pattern).
__device__ __forceinline__ void atomicMaxF(float* a, float v)
{
    if (v >= 0.f) atomicMax((int*)a, __float_as_int(v));
    else          atomicMin((unsigned int*)a, __float_as_uint(v));
}

// ---------------------------------------------------------------------------
// Pass 1 over edges: segment max of leaky-relu logits per (dst, head)
// ---------------------------------------------------------------------------
__global__ __launch_bounds__(256) void edge_max(const int* __restrict__ ei,
                                                const float* __restrict__ a_src,
                                                const float* __restrict__ a_dst,
                                                float* __restrict__ m)
{
    int e = blockIdx.x * 256 + threadIdx.x;
    if (e >= E_TOT) return;
    int s, d; edge_sd(ei, e, s, d);
    const float4* as = (const float4*)&a_src[(size_t)s * HEADS];
    const float4* ad = (const float4*)&a_dst[(size_t)d * HEADS];
    float4 s0 = as[0], s1 = as[1], d0 = ad[0], d1 = ad[1];
    float ev[8] = { s0.x + d0.x, s0.y + d0.y, s0.z + d0.z, s0.w + d0.w,
                    s1.x + d1.x, s1.y + d1.y, s1.z + d1.z, s1.w + d1.w };
    float* mb = &m[(size_t)d * HEADS];
#pragma unroll
    for (int hh = 0; hh < HEADS; ++hh) atomicMaxF(&mb[hh], lrelu(ev[hh]));
}

// ---------------------------------------------------------------------------
// Pass 2 over edges: denom = segment sum of exp(e - m[dst])
// ---------------------------------------------------------------------------
__global__ __launch_bounds__(256) void edge_sum(const int* __restrict__ ei,
                                                const float* __restrict__ a_src,
                                                const float* __restrict__ a_dst,
                                                const float* __restrict__ m,
                                                float* __restrict__ denom)
{
    int e = blockIdx.x * 256 + threadIdx.x;
    if (e >= E_TOT) return;
    int s, d; edge_sd(ei, e, s, d);
    const float4* as = (const float4*)&a_src[(size_t)s * HEADS];
    const float4* ad = (const float4*)&a_dst[(size_t)d * HEADS];
    const float4* mm = (const float4*)&m[(size_t)d * HEADS];
    float4 s0 = as[0], s1 = as[1], d0 = ad[0], d1 = ad[1];
    float4 m0 = mm[0], m1 = mm[1];
    float ev[8] = { s0.x + d0.x, s0.y + d0.y, s0.z + d0.z, s0.w + d0.w,
                    s1.x + d1.x, s1.y + d1.y, s1.z + d1.z, s1.w + d1.w };
    float mv[8] = { m0.x, m0.y, m0.z, m0.w, m1.x, m1.y, m1.z, m1.w };
    float* db = &denom[(size_t)d * HEADS];
#pragma unroll
    for (int hh = 0; hh < HEADS; ++hh)
        atomicAdd(&db[hh], __expf(lrelu(ev[hh]) - mv[hh]));
}

// ---------------------------------------------------------------------------
// Pass 3: one wave per edge. lane = output feature f (0..31).
// Heads are reduced in-register: lane hh (hh<8) computes alpha for head hh,
// broadcast via shfl; contribution = (1/8) * sum_h alpha_h * h[src,h,f],
// atomically added to out[dst,f]. This fuses per-head aggregation + head-mean.
// ---------------------------------------------------------------------------
__global__ __launch_bounds__(256) void edge_agg(const int* __restrict__ ei,
                                                const float* __restrict__ a_src,
                                                const float* __restrict__ a_dst,
                                                const float* __restrict__ m,
                                                const float* __restrict__ denom,
                                                const float* __restrict__ h,
                                                float* __restrict__ out)
{
    const int gt   = blockIdx.x * 256 + threadIdx.x;
    const int e    = gt >> 5;         // wave-uniform -> scalarized loads
    const int lane = gt & 31;
    if (e >= E_TOT) return;

    int s, d; edge_sd(ei, e, s, d);
    const int hd = lane & 7;
    const float ev    = lrelu(a_src[(size_t)s * HEADS + hd] +
                              a_dst[(size_t)d * HEADS + hd]);
    const float alpha = __expf(ev - m[(size_t)d * HEADS + hd]) /
                        (denom[(size_t)d * HEADS + hd] + 1e-16f);

    const float* hs = &h[(size_t)s * HF];
    float acc = 0.f;
#pragma unroll
    for (int hh = 0; hh < HEADS; ++hh) {
        float a = __shfl(alpha, hh, 32);
        acc += a * hs[hh * OUT_DIM + lane];   // coalesced 128B per head
    }
    atomicAdd(&out[(size_t)d * OUT_DIM + lane], acc * 0.125f);
}

// ---------------------------------------------------------------------------
// Launch
// ---------------------------------------------------------------------------
extern "C" void kernel_launch(void* const* d_in, const int* in_sizes, int n_in,
                              void* d_out, int out_size, void* d_ws, size_t ws_size,
                              hipStream_t stream)
{
    const float* X       = (const float*)d_in[0];   // [N,128]
    const int*   ei      = (const int*)  d_in[1];   // [2,E]
    const float* W       = (const float*)d_in[2];   // [128,256]
    const float* att_src = (const float*)d_in[3];   // [8,32]
    const float* att_dst = (const float*)d_in[4];   // [8,32]
    const float* bias    = (const float*)d_in[5];   // [32]
    float* out = (float*)d_out;                     // [N,32]

    char* ws = (char*)d_ws;
    float* h     = (float*)ws; ws += (size_t)N_NODES * HF    * sizeof(float); // 102.4 MB
    float* a_src = (float*)ws; ws += (size_t)N_NODES * HEADS * sizeof(float);
    float* a_dst = (float*)ws; ws += (size_t)N_NODES * HEADS * sizeof(float);
    float* m     = (float*)ws; ws += (size_t)N_NODES * HEADS * sizeof(float);
    float* denom = (float*)ws; ws += (size_t)N_NODES * HEADS * sizeof(float);

    init_bufs<<<(N_NODES * OUT_DIM + 255) / 256, 256, 0, stream>>>(m, denom, out, bias);
    gemm_h<<<N_NODES / 16, 256, 0, stream>>>(X, W, h);
    attn_logits<<<(N_NODES * HEADS + 255) / 256, 256, 0, stream>>>(h, att_src, att_dst,
                                                                   a_src, a_dst);
    edge_max<<<(E_TOT + 255) / 256, 256, 0, stream>>>(ei, a_src, a_dst, m);
    edge_sum<<<(E_TOT + 255) / 256, 256, 0, stream>>>(ei, a_src, a_dst, m, denom);
    edge_agg<<<((size_t)E_TOT * 32 + 255) / 256, 256, 0, stream>>>(ei, a_src, a_dst,
                                                                   m, denom, h, out);
}